// NodeEquilibriumLoss_15814069584181
// MI455X (gfx1250) — compile-verified
//
#include <hip/hip_runtime.h>

typedef __attribute__((ext_vector_type(2))) float v2f;
typedef __attribute__((ext_vector_type(8))) float v8f;

#define N_ELEM  2048
#define N_NODES 1024
#define RES_N   (N_NODES * 2)   // 2048, == N_ELEM
#define BLOCK   256
#define ROWS    4               // batch rows per workgroup (amortize incidence reads)

// One workgroup per ROWS batch rows.
//  - stage axial = EA*e for ROWS rows into LDS (gather locality for elem_ids)
//  - read each incidence entry ONCE, scatter into ROWS LDS residual images
//    via native ds_add_f32 atomics (cuts L2 incidence traffic by ROWS)
//  - fused (res - q - r)^2 streaming accumulate (prefetched during scatter)
//  - block reduction of 256 partials via V_WMMA_F32_16X16X4_F32 (ones x partials)
__global__ __launch_bounds__(BLOCK) void node_equil_main(
    const float* __restrict__ EA, const float* __restrict__ e,
    const float* __restrict__ q,  const float* __restrict__ r,
    const float* __restrict__ vecs, const int* __restrict__ node_ids,
    const int* __restrict__ elem_ids, float* __restrict__ partial,
    int e2, int Btotal)
{
    __shared__ float axial[ROWS][N_ELEM];   // 32 KB
    __shared__ float res[ROWS][RES_N];      // 32 KB
    __shared__ float sdata[BLOCK];          // 1 KB

    const int tid = threadIdx.x;
    const int b0  = blockIdx.x * ROWS;

    // ---- stage axial rows & zero residuals (vectorized b128 traffic) ----
#pragma unroll
    for (int row = 0; row < ROWS; ++row) {
        const int b = b0 + row;
        const bool valid = (b < Btotal);
        const float4* EA4 = (const float4*)(EA + (size_t)b * N_ELEM);
        const float4* e4  = (const float4*)(e  + (size_t)b * N_ELEM);
        float4* ax4  = (float4*)axial[row];
        float4* res4 = (float4*)res[row];
        for (int j = tid; j < N_ELEM / 4; j += BLOCK) {
            float4 p; p.x = 0.f; p.y = 0.f; p.z = 0.f; p.w = 0.f;
            if (valid) {
                float4 a = EA4[j];
                float4 s = e4[j];
                p.x = a.x * s.x; p.y = a.y * s.y; p.z = a.z * s.z; p.w = a.w * s.w;
            }
            ax4[j] = p;
            float4 z; z.x = 0.f; z.y = 0.f; z.z = 0.f; z.w = 0.f;
            res4[j] = z;                        // RES_N == N_ELEM: same trip count
        }
        // warm WGP/L2 for the post-barrier q/r stream (global_prefetch_b8)
        if (valid) {
            __builtin_prefetch(q + (size_t)b * RES_N + tid * 8, 0, 0);
            __builtin_prefetch(r + (size_t)b * RES_N + tid * 8, 0, 0);
        }
    }
    __syncthreads();

    // ---- gather / scale / scatter: each incidence entry read once ----
    const int4*   nid4 = (const int4*)node_ids;
    const int4*   eid4 = (const int4*)elem_ids;
    const float4* vv4  = (const float4*)vecs;   // 2 entries per float4
    const int e2q = e2 >> 2;
    for (int i4 = tid; i4 < e2q; i4 += BLOCK) {
        const int4   nn = nid4[i4];
        const int4   mm = eid4[i4];
        const float4 va = vv4[i4 * 2 + 0];      // entries 4*i4 + 0,1
        const float4 vb = vv4[i4 * 2 + 1];      // entries 4*i4 + 2,3
#pragma unroll
        for (int row = 0; row < ROWS; ++row) {
            const float a0 = axial[row][mm.x];
            const float a1 = axial[row][mm.y];
            const float a2 = axial[row][mm.z];
            const float a3 = axial[row][mm.w];
            atomicAdd(&res[row][2 * nn.x + 0], a0 * va.x);   // ds_add_f32
            atomicAdd(&res[row][2 * nn.x + 1], a0 * va.y);
            atomicAdd(&res[row][2 * nn.y + 0], a1 * va.z);
            atomicAdd(&res[row][2 * nn.y + 1], a1 * va.w);
            atomicAdd(&res[row][2 * nn.z + 0], a2 * vb.x);
            atomicAdd(&res[row][2 * nn.z + 1], a2 * vb.y);
            atomicAdd(&res[row][2 * nn.w + 0], a3 * vb.z);
            atomicAdd(&res[row][2 * nn.w + 1], a3 * vb.w);
        }
    }
    // scalar tail (e2 not multiple of 4)
    for (int i = (e2q << 2) + tid; i < e2; i += BLOCK) {
        const int m = elem_ids[i];
        const int n = node_ids[i];
        const float2 v = ((const float2*)vecs)[i];
#pragma unroll
        for (int row = 0; row < ROWS; ++row) {
            const float a = axial[row][m];
            atomicAdd(&res[row][2 * n + 0], a * v.x);
            atomicAdd(&res[row][2 * n + 1], a * v.y);
        }
    }
    __syncthreads();

    // ---- fused subtract + square + per-thread accumulate over ROWS rows ----
    float acc = 0.f;
#pragma unroll
    for (int row = 0; row < ROWS; ++row) {
        const int b = b0 + row;
        if (b >= Btotal) break;
        const float4* q4 = (const float4*)(q + (size_t)b * RES_N);
        const float4* r4 = (const float4*)(r + (size_t)b * RES_N);
        const float4* s4 = (const float4*)res[row];
        for (int j = tid; j < RES_N / 4; j += BLOCK) {
            float4 Q = q4[j];
            float4 R = r4[j];
            float4 S = s4[j];
            float t0 = S.x - Q.x - R.x;
            float t1 = S.y - Q.y - R.y;
            float t2 = S.z - Q.z - R.z;
            float t3 = S.w - Q.w - R.w;
            acc = fmaf(t0, t0, acc);
            acc = fmaf(t1, t1, acc);
            acc = fmaf(t2, t2, acc);
            acc = fmaf(t3, t3, acc);
        }
    }
    sdata[tid] = acc;
    __syncthreads();

    // ---- WMMA block reduction (wave 0 only; EXEC all-ones within the wave) ----
    // D = ones(16x4) x B(4x16) + C : every row of D holds the 16 column sums
    // of B, so summing row 0 over n gives the sum of all 64 B entries. Four
    // accumulating WMMAs fold all 256 partials exactly (x1.0 is exact, f32 RNE).
    if (tid < 32) {
        v2f ones; ones[0] = 1.0f; ones[1] = 1.0f;
        v8f d = {};
#pragma unroll
        for (int it = 0; it < 4; ++it) {
            v2f bm;
            bm[0] = sdata[it * 64 + tid];
            bm[1] = sdata[it * 64 + 32 + tid];
            d = __builtin_amdgcn_wmma_f32_16x16x4_f32(
                    /*neg_a=*/false, ones, /*neg_b=*/false, bm,
                    /*c_mod=*/(short)0, d, /*reuse_a=*/false, /*reuse_b=*/false);
        }
        // d[0]: lanes 0-15 = D[0,n], lanes 16-31 = D[8,n] (identical rows)
        // -> cross-lane sum of d[0] equals 2 * total
        float x = d[0];
        for (int off = 16; off > 0; off >>= 1)
            x += __shfl_down(x, off, 32);
        if (tid == 0) partial[blockIdx.x] = x * 0.5f;
    }
}

// Deterministic final reduction: sum 'nb' per-block partials, scale to mean.
__global__ __launch_bounds__(BLOCK) void node_equil_reduce(
    const float* __restrict__ partial, float* __restrict__ out,
    int nb, float inv_count)
{
    __shared__ float sdata[BLOCK];
    float acc = 0.f;
    for (int i = threadIdx.x; i < nb; i += BLOCK) acc += partial[i];
    sdata[threadIdx.x] = acc;
    __syncthreads();
    for (int s = BLOCK / 2; s > 0; s >>= 1) {
        if (threadIdx.x < s) sdata[threadIdx.x] += sdata[threadIdx.x + s];
        __syncthreads();
    }
    if (threadIdx.x == 0) out[0] = sdata[0] * inv_count;
}

extern "C" void kernel_launch(void* const* d_in, const int* in_sizes, int n_in,
                              void* d_out, int out_size, void* d_ws, size_t ws_size,
                              hipStream_t stream) {
    const float* EA       = (const float*)d_in[0];
    const float* e        = (const float*)d_in[1];
    const float* q        = (const float*)d_in[2];
    const float* r        = (const float*)d_in[3];
    const float* vecs     = (const float*)d_in[4];
    const int*   node_ids = (const int*)d_in[5];
    const int*   elem_ids = (const int*)d_in[6];

    const int B  = in_sizes[0] / N_ELEM;          // 4096
    const int e2 = in_sizes[5];                   // 4096
    const int nblocks = (B + ROWS - 1) / ROWS;    // 1024

    float* partial = (float*)d_ws;                // nblocks floats of scratch
    float* out     = (float*)d_out;

    node_equil_main<<<dim3(nblocks), dim3(BLOCK), 0, stream>>>(
        EA, e, q, r, vecs, node_ids, elem_ids, partial, e2, B);

    const float inv_count = 1.0f / ((float)B * (float)RES_N);
    node_equil_reduce<<<dim3(1), dim3(BLOCK), 0, stream>>>(partial, out, nblocks, inv_count);
}